// MemNet_13778255086130
// MI455X (gfx1250) — compile-verified
//
#include <hip/hip_runtime.h>

// Problem constants (match reference)
#define BB        64
#define OBS_LEN   512
#define MEM_LEN   512
#define ROW_LEN   1024     // OBS_LEN + MEM_LEN
#define N_MEMS    4096
#define EMBED     64
#define HEADS     4
#define KSEL      10
#define HEAD_DIM  16

#define NBANDS    32       // row-bands for the column-stat partial reduction
#define BAND_ROWS (N_MEMS / NBANDS)   // 128 rows per band

typedef float v2f __attribute__((ext_vector_type(2)));
typedef float v8f __attribute__((ext_vector_type(8)));

// D = A(16x4) * B(4x16) + C, fp32 WMMA (wave32). 8-arg form:
// (neg_a, A, neg_b, B, c_mod, C, reuse_a, reuse_b)
__device__ __forceinline__ v8f wmma4(v2f a, v2f b, v8f c) {
    return __builtin_amdgcn_wmma_f32_16x16x4_f32(false, a, false, b, (short)0, c,
                                                 false, false);
}

// gfx1250 async global->LDS copy (GV mode): VDST = LDS byte address,
// VADDR = 64-bit global address. Tracked by ASYNCcnt.
__device__ __forceinline__ void async_load_b128_to_lds(unsigned lds_byte_addr,
                                                       const void* gaddr) {
    asm volatile("global_load_async_to_lds_b128 %0, %1, off"
                 :
                 : "v"(lds_byte_addr), "v"(gaddr)
                 : "memory");
}
__device__ __forceinline__ void wait_asynccnt0() {
    asm volatile("s_wait_asynccnt 0" ::: "memory");
}

// ---------------------------------------------------------------------------
// Kernel 1a: per-row-band partial column sums over memories[:, :OBS_LEN].
// Fully coalesced: lane t reads columns t and t+256 of each row.
// Deterministic (serial per-thread accumulation, fixed order).
// ---------------------------------------------------------------------------
__global__ __launch_bounds__(256)
void k_colstats_partial(const float* __restrict__ memories,
                        float* __restrict__ P1, float* __restrict__ P2) {
    const int g = blockIdx.x;            // band index, NBANDS bands
    const int t = threadIdx.x;
    const int j0 = t, j1 = t + 256;
    float a1 = 0.f, a2 = 0.f, b1 = 0.f, b2 = 0.f;
    const float* base = memories + (size_t)g * BAND_ROWS * ROW_LEN;
    for (int r = 0; r < BAND_ROWS; ++r) {
        const float* row = base + (size_t)r * ROW_LEN;
        __builtin_prefetch(row + ROW_LEN, 0, 1);   // global_prefetch_b8
        const float v0 = row[j0];
        const float v1 = row[j1];
        a1 += v0; a2 += v0 * v0;
        b1 += v1; b2 += v1 * v1;
    }
    P1[g * OBS_LEN + j0] = a1;  P2[g * OBS_LEN + j0] = a2;
    P1[g * OBS_LEN + j1] = b1;  P2[g * OBS_LEN + j1] = b2;
}

// ---------------------------------------------------------------------------
// Kernel 1b: reduce the NBANDS partials per column (fixed serial order).
// ---------------------------------------------------------------------------
__global__ __launch_bounds__(256)
void k_colstats_reduce(const float* __restrict__ P1, const float* __restrict__ P2,
                       float* __restrict__ S1, float* __restrict__ S2) {
    const int j = blockIdx.x * 256 + threadIdx.x;   // 2 blocks x 256 = 512 cols
    float s1 = 0.f, s2 = 0.f;
    for (int g = 0; g < NBANDS; ++g) {
        s1 += P1[g * OBS_LEN + j];
        s2 += P2[g * OBS_LEN + j];
    }
    S1[j] = s1;
    S2[j] = s2;
}

// ---------------------------------------------------------------------------
// Kernel 2: per batch row, dist^2[j] = S2[j] - 2*obs*S1[j] + N_MEMS*obs^2,
// then select the 10 smallest (tie-break: lowest index, matching top_k).
// Key pack: dist2 is >=0 so its IEEE bits are order-preserving as u32.
// ---------------------------------------------------------------------------
__global__ __launch_bounds__(512)
void k_topk(const float* __restrict__ obs, const float* __restrict__ S1,
            const float* __restrict__ S2, int* __restrict__ idxout) {
    const int b = blockIdx.x;
    const int t = threadIdx.x;                  // 512 threads = 512 columns
    const float o  = obs[b * OBS_LEN + t];
    const float d2 = S2[t] - 2.f * o * S1[t] + (float)N_MEMS * o * o;
    unsigned long long key =
        ((unsigned long long)__float_as_uint(d2) << 32) | (unsigned)t;

    __shared__ unsigned long long sk[512];
    __shared__ unsigned long long best;
    for (int it = 0; it < KSEL; ++it) {
        sk[t] = key;
        __syncthreads();
        for (int s = 256; s > 0; s >>= 1) {
            if (t < s) {
                unsigned long long o2 = sk[t + s];
                if (o2 < sk[t]) sk[t] = o2;
            }
            __syncthreads();
        }
        if (t == 0) {
            best = sk[0];
            idxout[b * KSEL + it] = (int)(sk[0] & 0xffffffffULL);
        }
        __syncthreads();
        if (key == best) key = 0xffffffffffffffffULL;  // remove winner
        __syncthreads();
    }
}

// ---------------------------------------------------------------------------
// WMMA fragment helpers. Layouts per ISA 7.12.2 (wave32):
//   A 16x4 f32:  a[j] = A[lane&15][2*(lane>>4)+j]
//   B 4x16 f32:  b[j] = B[2*(lane>>4)+j][lane&15]   (rows striped like A's K)
//   C/D 16x16:   c[r] = C[8*(lane>>4)+r][lane&15]
// B is taken from a row-major weight W (N x Ktot) as B[k][n] = W[n*ldw + k].
// ---------------------------------------------------------------------------
__device__ __forceinline__ v2f frag_rowmajor(const float* __restrict__ row,
                                             int kk, int half) {
    v2f f;
    f[0] = row[kk + half * 2 + 0];
    f[1] = row[kk + half * 2 + 1];
    return f;
}

// ---------------------------------------------------------------------------
// Kernel 3: OE = obs(64x512) @ w_obs^T + b_obs  (K=512)  -> LDS
//           Q  = OE(64x64)   @ wq^T    + bq     (K=64)   -> global ws
// One 512-thread block: 16 waves, one 16x16 tile per wave.
// ---------------------------------------------------------------------------
__global__ __launch_bounds__(512)
void k_embed_q(const float* __restrict__ obs,
               const float* __restrict__ w_obs, const float* __restrict__ b_obs,
               const float* __restrict__ wq,    const float* __restrict__ bq,
               float* __restrict__ Q) {
    __shared__ float OE[EMBED * EMBED];
    const int t = threadIdx.x;
    const int wave = t >> 5, lane = t & 31;
    const int col = lane & 15, half = lane >> 4;
    const int mt = wave >> 2, nt = wave & 3;

    // Stage 1: OE tile (mt,nt), K = 512
    v8f acc = {0, 0, 0, 0, 0, 0, 0, 0};
    const float* arow = obs   + (size_t)(mt * 16 + col) * OBS_LEN;
    const float* brow = w_obs + (size_t)(nt * 16 + col) * OBS_LEN;
    for (int kk = 0; kk < OBS_LEN; kk += 4)
        acc = wmma4(frag_rowmajor(arow, kk, half),
                    frag_rowmajor(brow, kk, half), acc);
    {
        const float bias = b_obs[nt * 16 + col];
        for (int r = 0; r < 8; ++r)
            OE[(mt * 16 + half * 8 + r) * EMBED + nt * 16 + col] = acc[r] + bias;
    }
    __syncthreads();

    // Stage 2: Q tile (mt,nt), K = 64, A from LDS
    v8f acc2 = {0, 0, 0, 0, 0, 0, 0, 0};
    const float* arow2 = &OE[(mt * 16 + col) * EMBED];
    const float* brow2 = wq + (size_t)(nt * 16 + col) * EMBED;
    for (int kk = 0; kk < EMBED; kk += 4)
        acc2 = wmma4(frag_rowmajor(arow2, kk, half),
                     frag_rowmajor(brow2, kk, half), acc2);
    {
        const float bias = bq[nt * 16 + col];
        for (int r = 0; r < 8; ++r)
            Q[(mt * 16 + half * 8 + r) * EMBED + nt * 16 + col] = acc2[r] + bias;
    }
}

// ---------------------------------------------------------------------------
// Kernel 4 (per batch b): async-gather 10 memory rows -> LDS (padded 16x512),
// K = mems@wk^T+bk, V = mems@wv^T+bv via WMMA (waves 0-3: K, 4-7: V),
// then 10-key softmax attention per head -> FEAT[b] (64 floats).
// ---------------------------------------------------------------------------
__global__ __launch_bounds__(256)
void k_kv_attn(const float* __restrict__ memories, const int* __restrict__ idxbuf,
               const float* __restrict__ Q,
               const float* __restrict__ wk, const float* __restrict__ bk,
               const float* __restrict__ wv, const float* __restrict__ bv,
               float* __restrict__ FEAT) {
    __shared__ __align__(16) float Ash[16 * MEM_LEN]; // 32 KB gathered mems
    __shared__ float Ks[16 * EMBED];
    __shared__ float Vs[16 * EMBED];
    const int b = blockIdx.x;
    const int t = threadIdx.x;

    // Async gather: each lane moves 16B per issue; 128 lanes cover one 2KB row,
    // so 256 lanes x 5 issues cover the 10 selected rows.
    {
        const int sub = t & 127;       // 16B chunk index within a row
        const int rr  = t >> 7;        // 0 or 1
        for (int i = 0; i < 5; ++i) {
            const int r = i * 2 + rr;  // rows 0..9
            const float* src = memories
                + (size_t)idxbuf[b * KSEL + r] * ROW_LEN + OBS_LEN + sub * 4;
            const unsigned ldsa =
                (unsigned)(uintptr_t)&Ash[r * MEM_LEN + sub * 4];
            async_load_b128_to_lds(ldsa, (const void*)src);
        }
        // zero-pad rows 10..15 (keeps WMMA A-tile rows 10..15 well-defined)
        for (int j = t; j < 6 * MEM_LEN; j += 256)
            Ash[10 * MEM_LEN + j] = 0.f;
        wait_asynccnt0();
    }
    __syncthreads();

    const int wave = t >> 5, lane = t & 31;
    const int col = lane & 15, half = lane >> 4;
    const int nt = wave & 3;
    const bool isV = wave >= 4;
    const float* W    = isV ? wv : wk;
    const float* bias = isV ? bv : bk;
    float*       dst  = isV ? Vs : Ks;

    v8f acc = {0, 0, 0, 0, 0, 0, 0, 0};
    const float* arow = &Ash[col * MEM_LEN];                     // M=16 tile, row=col
    const float* brow = W + (size_t)(nt * 16 + col) * MEM_LEN;
    for (int kk = 0; kk < MEM_LEN; kk += 4)
        acc = wmma4(frag_rowmajor(arow, kk, half),
                    frag_rowmajor(brow, kk, half), acc);
    {
        const float bb2 = bias[nt * 16 + col];
        for (int r = 0; r < 8; ++r)
            dst[(half * 8 + r) * EMBED + nt * 16 + col] = acc[r] + bb2;
    }
    __syncthreads();

    // Attention: thread e (<64) -> head h = e/16, dim d = e%16
    if (t < EMBED) {
        const int h = t >> 4, d = t & 15;
        const float* qrow = Q + b * EMBED + h * HEAD_DIM;
        float lg[KSEL];
        float mx = -3.402823466e38f;
        for (int k2 = 0; k2 < KSEL; ++k2) {
            float s = 0.f;
            for (int dd = 0; dd < HEAD_DIM; ++dd)
                s += qrow[dd] * Ks[k2 * EMBED + h * HEAD_DIM + dd];
            s *= 0.25f;                 // 1/sqrt(HEAD_DIM=16)
            lg[k2] = s;
            mx = fmaxf(mx, s);
        }
        float sum = 0.f;
        for (int k2 = 0; k2 < KSEL; ++k2) { lg[k2] = __expf(lg[k2] - mx); sum += lg[k2]; }
        const float inv = 1.f / sum;
        float f = 0.f;
        for (int k2 = 0; k2 < KSEL; ++k2)
            f += lg[k2] * inv * Vs[k2 * EMBED + h * HEAD_DIM + d];
        FEAT[b * EMBED + t] = f;
    }
}

// ---------------------------------------------------------------------------
// Kernel 5: O1 = FEAT(64x64)@wo^T + bo -> LDS; logits = O1@w_out^T + b_out.
// ---------------------------------------------------------------------------
__global__ __launch_bounds__(512)
void k_out(const float* __restrict__ FEAT,
           const float* __restrict__ wo,    const float* __restrict__ bo,
           const float* __restrict__ w_out, const float* __restrict__ b_out,
           float* __restrict__ out) {
    __shared__ float O1[BB * EMBED];
    const int t = threadIdx.x;
    const int wave = t >> 5, lane = t & 31;
    const int col = lane & 15, half = lane >> 4;
    const int mt = wave >> 2, nt = wave & 3;

    v8f acc = {0, 0, 0, 0, 0, 0, 0, 0};
    const float* arow = FEAT + (size_t)(mt * 16 + col) * EMBED;
    const float* brow = wo   + (size_t)(nt * 16 + col) * EMBED;
    for (int kk = 0; kk < EMBED; kk += 4)
        acc = wmma4(frag_rowmajor(arow, kk, half),
                    frag_rowmajor(brow, kk, half), acc);
    {
        const float bias = bo[nt * 16 + col];
        for (int r = 0; r < 8; ++r)
            O1[(mt * 16 + half * 8 + r) * EMBED + nt * 16 + col] = acc[r] + bias;
    }
    __syncthreads();

    v8f acc2 = {0, 0, 0, 0, 0, 0, 0, 0};
    const float* arow2 = &O1[(mt * 16 + col) * EMBED];
    const float* brow2 = w_out + (size_t)(nt * 16 + col) * EMBED;
    for (int kk = 0; kk < EMBED; kk += 4)
        acc2 = wmma4(frag_rowmajor(arow2, kk, half),
                     frag_rowmajor(brow2, kk, half), acc2);
    {
        const float bias = b_out[nt * 16 + col];
        for (int r = 0; r < 8; ++r)
            out[(mt * 16 + half * 8 + r) * EMBED + nt * 16 + col] = acc2[r] + bias;
    }
}

// ---------------------------------------------------------------------------
extern "C" void kernel_launch(void* const* d_in, const int* in_sizes, int n_in,
                              void* d_out, int out_size, void* d_ws, size_t ws_size,
                              hipStream_t stream) {
    (void)in_sizes; (void)n_in; (void)out_size; (void)ws_size;
    const float* obs      = (const float*)d_in[0];
    const float* memories = (const float*)d_in[1];
    const float* w_obs    = (const float*)d_in[2];
    const float* b_obs    = (const float*)d_in[3];
    const float* wq       = (const float*)d_in[4];
    const float* bq       = (const float*)d_in[5];
    const float* wk       = (const float*)d_in[6];
    const float* bk       = (const float*)d_in[7];
    const float* wv       = (const float*)d_in[8];
    const float* bv       = (const float*)d_in[9];
    const float* wo       = (const float*)d_in[10];
    const float* bo       = (const float*)d_in[11];
    const float* w_out    = (const float*)d_in[12];
    const float* b_out    = (const float*)d_in[13];

    // Workspace layout (floats):
    // S1[512] | S2[512] | idx[640 ints] | Q[4096] | FEAT[4096] |
    // P1[NBANDS*512] | P2[NBANDS*512]
    float* wsf   = (float*)d_ws;
    float* S1    = wsf;
    float* S2    = wsf + 512;
    int*   idxb  = (int*)(wsf + 1024);
    float* Q     = wsf + 1024 + BB * KSEL;
    float* FEAT  = Q + BB * EMBED;
    float* P1    = FEAT + BB * EMBED;
    float* P2    = P1 + NBANDS * OBS_LEN;

    k_colstats_partial<<<NBANDS, 256, 0, stream>>>(memories, P1, P2);
    k_colstats_reduce <<<2, 256, 0, stream>>>(P1, P2, S1, S2);
    k_topk    <<<BB, 512, 0, stream>>>(obs, S1, S2, idxb);
    k_embed_q <<<1, 512, 0, stream>>>(obs, w_obs, b_obs, wq, bq, Q);
    k_kv_attn <<<BB, 256, 0, stream>>>(memories, idxb, Q, wk, bk, wv, bv, FEAT);
    k_out     <<<1, 512, 0, stream>>>(FEAT, wo, bo, w_out, b_out, (float*)d_out);
}